// QuantumLayer_1717986918930
// MI455X (gfx1250) — compile-verified
//
#include <hip/hip_runtime.h>

typedef __attribute__((ext_vector_type(2))) float v2f;
typedef __attribute__((ext_vector_type(8))) float v8f;
typedef __attribute__((ext_vector_type(4))) unsigned v4u;
typedef __attribute__((ext_vector_type(8))) unsigned v8u;

#define NMODES   12
#define NPHOT    6
#define NSTATES  924
#define PSTR     928           // probs row stride, padded: 928 = 4 waves * 58 steps * 4
#define NTHETA   120
#define NPHASES  132
#define BATCH    64
#define OUTD     10
#define TWO_PI   6.283185307179586f

struct cplx { float x, y; };
__device__ __forceinline__ cplx cmul(cplx a, cplx b) {
  cplx r;
  r.x = fmaf(a.x, b.x, -a.y * b.y);
  r.y = fmaf(a.x, b.y,  a.y * b.x);
  return r;
}

// BINOM[n][k] = C(n,k), n<=11, k<=5 (for combination unranking)
__device__ __constant__ int BINOM[12][6] = {
  {1,0,0,0,0,0},{1,1,0,0,0,0},{1,2,1,0,0,0},{1,3,3,1,0,0},
  {1,4,6,4,1,0},{1,5,10,10,5,1},{1,6,15,20,15,6},{1,7,21,35,35,21},
  {1,8,28,56,70,56},{1,9,36,84,126,126},{1,10,45,120,210,252},
  {1,11,55,165,330,462}};

// ---------------------------------------------------------------------------
// Kernel 1: one workgroup per batch element.
//   - build 12x12 unitary in LDS via 12 Clements layers (parallel within layer)
//   - 924 permanents of 6x6 submatrices via fully-unrolled Gray-code Ryser
//   - normalize (LDS atomic reduction), write probs[b][928] (4 zero pad cols)
// ---------------------------------------------------------------------------
__global__ __launch_bounds__(256) void qlayer_probs(
    const float* __restrict__ x,        // [64,12]
    const float* __restrict__ thetas,   // [120]
    float* __restrict__ probs_out)      // [64,928]
{
  __shared__ float ph[NPHASES];
  __shared__ cplx  Uld[NMODES][NMODES];
  __shared__ float probs_s[NSTATES];
  __shared__ float s_total;

  const int b   = blockIdx.x;
  const int tid = threadIdx.x;

  if (tid < NTHETA)       ph[tid] = thetas[tid];
  else if (tid < NPHASES) ph[tid] = TWO_PI * x[b * (NPHASES - NTHETA) + (tid - NTHETA)];
  if (tid < NMODES * NMODES) {
    cplx e; e.x = (tid / NMODES == tid % NMODES) ? 1.0f : 0.0f; e.y = 0.0f;
    Uld[tid / NMODES][tid % NMODES] = e;
  }
  if (tid == 0) s_total = 0.0f;
  __syncthreads();

  // ---- Clements mesh: 12 layers; pairs within a layer are disjoint ----
  for (int l = 0; l < NMODES; ++l) {
    const int odd    = l & 1;
    const int npairs = odd ? 5 : 6;
    const int base   = 6 * ((l + 1) >> 1) + 5 * (l >> 1);  // MZIs before layer l
    const int nwork  = npairs * NMODES;
    if (tid < nwork) {
      const int pi = tid / NMODES;
      const int c  = tid % NMODES;
      const int p  = odd + 2 * pi;
      const int k  = base + pi;
      const float f1 = ph[2 * k];
      const float f2 = ph[2 * k + 1];
      // T' = BS*PS(f2)*BS = i*e^{i f2/2} [[s, c],[c, -s]],  s=sin(f2/2), c=cos(f2/2)
      const float h  = 0.5f * f2;
      const float sh = __sinf(h), ch = __cosf(h);
      cplx w;  w.x = -sh; w.y = ch;                 // i * e^{i h}
      cplx T00, T01, T10, T11;
      T00.x = w.x * sh; T00.y = w.y * sh;
      T01.x = w.x * ch; T01.y = w.y * ch;
      T10 = T01;
      T11.x = -T00.x; T11.y = -T00.y;
      cplx e1; e1.x = __cosf(f1); e1.y = __sinf(f1);  // column 0 *= e^{i f1}
      T00 = cmul(T00, e1);
      T10 = cmul(T10, e1);

      const cplx u0 = Uld[p][c];
      const cplx u1 = Uld[p + 1][c];
      cplx n0, n1;
      n0.x = T00.x*u0.x - T00.y*u0.y + T01.x*u1.x - T01.y*u1.y;
      n0.y = T00.x*u0.y + T00.y*u0.x + T01.x*u1.y + T01.y*u1.x;
      n1.x = T10.x*u0.x - T10.y*u0.y + T11.x*u1.x - T11.y*u1.y;
      n1.y = T10.x*u0.y + T10.y*u0.x + T11.x*u1.y + T11.y*u1.x;
      Uld[p][c]     = n0;
      Uld[p + 1][c] = n1;
    }
    __syncthreads();
  }

  // ---- permanents via Ryser (Gray code), columns 0..5 fixed ----
  float lsum = 0.0f;
  #pragma unroll 1
  for (int o = tid; o < NSTATES; o += 256) {
    // unrank combination o of C(12,6) in lex order
    int modes[NPHOT];
    {
      int r = o, v = 0;
      #pragma unroll
      for (int i = 0; i < NPHOT; ++i) {
        while (true) {
          const int cnt = BINOM[11 - v][5 - i];
          if (r < cnt) break;
          r -= cnt; ++v;
        }
        modes[i] = v; ++v;
      }
    }
    cplx A[NPHOT][NPHOT];
    #pragma unroll
    for (int i = 0; i < NPHOT; ++i)
      #pragma unroll
      for (int j = 0; j < NPHOT; ++j)
        A[i][j] = Uld[modes[i]][j];

    cplx rs[NPHOT];
    #pragma unroll
    for (int i = 0; i < NPHOT; ++i) { rs[i].x = 0.0f; rs[i].y = 0.0f; }
    float pr = 0.0f, pim = 0.0f;
    unsigned prev = 0u;
    #pragma unroll
    for (unsigned g = 1; g < 64u; ++g) {
      const unsigned gray = g ^ (g >> 1);
      const unsigned diff = gray ^ prev;          // single bit; constant after unroll
      prev = gray;
      const int   j  = __builtin_ctz(diff);
      const float sb = (gray & diff) ? 1.0f : -1.0f;
      #pragma unroll
      for (int i = 0; i < NPHOT; ++i) {
        rs[i].x = fmaf(sb, A[i][j].x, rs[i].x);
        rs[i].y = fmaf(sb, A[i][j].y, rs[i].y);
      }
      cplx p = rs[0];
      #pragma unroll
      for (int i = 1; i < NPHOT; ++i) p = cmul(p, rs[i]);
      const float sg = (g & 1u) ? -1.0f : 1.0f;   // (-1)^6 * (-1)^{|S|}
      pr  = fmaf(sg, p.x, pr);
      pim = fmaf(sg, p.y, pim);
    }
    const float prob = pr * pr + pim * pim;
    probs_s[o] = prob;
    lsum += prob;
  }
  atomicAdd(&s_total, lsum);          // ds_add_f32
  __syncthreads();

  const float inv = 1.0f / s_total;
  for (int o = tid; o < PSTR; o += 256) {
    const float v = (o < NSTATES) ? probs_s[o] * inv : 0.0f;  // zero K pad
    probs_out[b * PSTR + o] = v;
  }
}

// ---------------------------------------------------------------------------
// Kernel 2: out[64,10] = probs[64,924] @ weight^T[924,10] + bias
//   - 4 blocks (one 16-row M tile each) x 4 waves (K split: 58 wmma each)
//   - weight [10,924] f32 (36,960 B) DMA'd to LDS once per block via the
//     Tensor Data Mover (tensor_load_to_lds, 2D descriptor), B reads = ds_load
//   - V_WMMA_F32_16X16X4_F32, constant trip count (fully unrollable, EXEC
//     untouched), partial tiles reduced via ds_add_f32
// ---------------------------------------------------------------------------
__global__ __launch_bounds__(128) void qlayer_linear(
    const float* __restrict__ probs,   // [64,928] padded
    const float* __restrict__ weight,  // [10,924] row-major
    const float* __restrict__ bias,    // [10]
    float* __restrict__ out)           // [64,10]
{
  __shared__ float wlds[OUTD * NSTATES + 16];  // +16: zero tail guard (no NaN)
  __shared__ float cred[16][16];               // cross-wave K reduction

  const int tid   = threadIdx.x;
  const int wave  = tid >> 5;          // 0..3 -> K chunk
  const int lane  = tid & 31;
  const int m0    = blockIdx.x * 16;   // M tile
  const int mrow  = m0 + (lane & 15);  // A-fragment row
  const int khalf = lane >> 4;         // A/B K sub-offset
  const int n     = lane & 15;         // B-fragment / output column
  const float nmask = (n < OUTD) ? 1.0f : 0.0f;
  const int   nc    = (n < OUTD) ? n : 0;

  // zero the reduction tile (128 threads cover 2 entries each) + LDS tail guard
  ((float*)cred)[tid]       = 0.0f;
  ((float*)cred)[tid + 128] = 0.0f;
  if (tid < 16) wlds[OUTD * NSTATES + tid] = 0.0f;

  // ---- TDM: DMA weight[10][924] (f32) into LDS; wave 0 issues ----
  if (wave == 0) {
    const unsigned lds_off = (unsigned)(unsigned long long)&wlds[0]; // LDS byte offset
    const unsigned long long ga = (unsigned long long)weight;
    // D# group0: count=1 | lds_addr | global_addr(57b) | type=2
    v4u g0;
    g0.s0 = 1u;                                   // count=1, user mode
    g0.s1 = lds_off;                              // lds_addr
    g0.s2 = (unsigned)ga;                         // global_addr[31:0]
    g0.s3 = (unsigned)((ga >> 32) & 0x1FFFFFFull) | (2u << 30); // [56:32], type=2
    // D# group1: data_size=4B, tensor 924x10, tile 924x10, stride0=924
    v8u g1;
    g1.s0 = (2u << 16);                            // workgroup_mask=0, data_size=2(4B)
    g1.s1 = (NSTATES & 0xFFFFu) << 16;             // tensor_dim0[15:0] @ bits63:48
    g1.s2 = (NSTATES >> 16) | ((unsigned)OUTD << 16); // tensor_dim0[31:16] | tensor_dim1[15:0]
    g1.s3 = ((unsigned)OUTD >> 16) | ((unsigned)NSTATES << 16); // tensor_dim1[31:16] | tile_dim0
    g1.s4 = (unsigned)OUTD;                        // tile_dim1=10, tile_dim2=0
    g1.s5 = (unsigned)NSTATES;                     // tensor_dim0_stride[31:0]
    g1.s6 = 0u;                                    // stride0[47:32] | stride1[15:0]
    g1.s7 = 0u;                                    // stride1[47:16]
    asm volatile("tensor_load_to_lds %0, %1"
                 :: "s"(g0), "s"(g1)
                 : "memory");
    __builtin_amdgcn_s_wait_tensorcnt(0);          // TENSORcnt==0: tile in LDS
  }
  __syncthreads();                                  // publish LDS tile + zeros

  // ---- K-split WMMA: wave w handles global k in [232w, 232w+232) ----
  // Constant 58-step trip count for every wave: no divergent exit, EXEC all-1s.
  // Pad columns k=924..927 have A==0, so B values there contribute exactly 0
  // (tail guard keeps them finite).
  const int kbase = wave * 232;                     // 58 steps * 4
  const float* __restrict__ arow = probs + mrow * PSTR + kbase + 2 * khalf;
  const float* __restrict__ wrow = wlds + nc * NSTATES + kbase + 2 * khalf; // LDS

  v8f acc = {};
  #pragma unroll
  for (int s = 0; s < 58; ++s) {
    const int k = s * 4;
    v2f a, bb;
    a.x  = arow[k];
    a.y  = arow[k + 1];
    bb.x = wrow[k]     * nmask;
    bb.y = wrow[k + 1] * nmask;
    acc = __builtin_amdgcn_wmma_f32_16x16x4_f32(
        /*neg_a=*/false, a, /*neg_b=*/false, bb,
        /*c_mod=*/(short)0, acc, /*reuse_a=*/false, /*reuse_b=*/false);
  }

  // ---- reduce the 4 partial tiles (masked columns are exactly zero) ----
  #pragma unroll
  for (int r = 0; r < 8; ++r)
    atomicAdd(&cred[r + 8 * khalf][n], acc[r]);     // ds_add_f32
  __syncthreads();

  if (tid < 16 * OUTD) {
    const int ml = tid / OUTD, nn = tid % OUTD;
    out[(m0 + ml) * OUTD + nn] = cred[ml][nn] + bias[nn];
  }
}

extern "C" void kernel_launch(void* const* d_in, const int* in_sizes, int n_in,
                              void* d_out, int out_size, void* d_ws, size_t ws_size,
                              hipStream_t stream) {
  const float* x      = (const float*)d_in[0];   // [64,12]
  const float* thetas = (const float*)d_in[1];   // [120]
  const float* weight = (const float*)d_in[2];   // [10,924]
  const float* bias   = (const float*)d_in[3];   // [10]
  float* out   = (float*)d_out;                  // [64,10]
  float* probs = (float*)d_ws;                   // [64,928] scratch (237,568 B)

  hipLaunchKernelGGL(qlayer_probs,  dim3(BATCH), dim3(256), 0, stream, x, thetas, probs);
  hipLaunchKernelGGL(qlayer_linear, dim3(4),     dim3(128), 0, stream, probs, weight, bias, out);
}